// LightGCN_54460185313775
// MI455X (gfx1250) — compile-verified
//
#include <hip/hip_runtime.h>
#include <stdint.h>

#define NUM_USERS 120000
#define NUM_ITEMS 30000
#define NUM_NODES (NUM_USERS + NUM_ITEMS)
#define EMBED_DIM 64
#define N_LAYERS  3
#define N_EDGES   4000000

#define TILE 2048   // edges per workgroup tile (24 KB LDS: col/row/w)

typedef unsigned int u32x4 __attribute__((ext_vector_type(4)));
typedef int          i32x8 __attribute__((ext_vector_type(8)));
typedef int          i32x4 __attribute__((ext_vector_type(4)));

// ---------------------------------------------------------------------------
// TDM: 1-D DMA of `tile_elems` 4-byte elements from global `gptr` into LDS
// byte offset `lds_off`. `n_valid` = remaining valid elements from gptr; tile
// reads past it return zero (hardware OOB), so the tail needs no special path.
// D# layout per CDNA5 ISA. This toolchain uses
// the 6-arg builtin: (g0 u32x4, g1 i32x8, g2 i32x4, g3 i32x4, ext i32x8, cpol).
// ---------------------------------------------------------------------------
__device__ __forceinline__ void tdm_load_1d_b32(unsigned lds_off, const void* gptr,
                                                unsigned n_valid, unsigned tile_elems) {
  uint64_t ga = (uint64_t)(uintptr_t)gptr;
  u32x4 g0;
  g0[0] = 1u;                                            // count=1 (user descriptor)
  g0[1] = lds_off;                                       // lds_addr (bytes)
  g0[2] = (unsigned)(ga & 0xFFFFFFFFu);                  // global_addr[31:0]
  g0[3] = (unsigned)((ga >> 32) & 0x01FFFFFFu)           // global_addr[56:32]
        | (2u << 30);                                    // type = 2 ("image")
  i32x8 g1;
  g1[0] = (int)(2u << 16);                               // wg_mask=0, data_size=2 (4B)
  g1[1] = (int)((n_valid & 0xFFFFu) << 16);              // tensor_dim0[15:0]
  g1[2] = (int)(((n_valid >> 16) & 0xFFFFu)              // tensor_dim0[31:16]
        | (1u << 16));                                   // tensor_dim1 = 1
  g1[3] = (int)((tile_elems & 0xFFFFu) << 16);           // tensor_dim1 hi=0 | tile_dim0
  g1[4] = 1;                                             // tile_dim1=1, tile_dim2=0
  g1[5] = (int)n_valid;                                  // tensor_dim0_stride[31:0]
  g1[6] = 0;                                             // stride0 hi | stride1 lo
  g1[7] = 0;                                             // stride1 hi
  i32x4 z4 = {0, 0, 0, 0};                               // groups 2/3 unused (<=2D)
  i32x8 z8 = {0, 0, 0, 0, 0, 0, 0, 0};                   // extended group (unused)
  __builtin_amdgcn_tensor_load_to_lds(g0, g1, z4, z4, z8, 0);
}

// ---------------------------------------------------------------------------
// Propagate: y[row[e]] += w[e] * x[col[e]]   (scatter-add SpMM, 64-wide rows)
// One block = one tile of TILE edges. Wave 0 DMAs the edge tile into LDS via
// the TDM, waits on TENSORcnt, then 16-lane groups process one edge each:
// lane k gathers float4 chunk k of x[col] (coalesced global_load_b128) and
// issues 4 non-returning f32 atomics into y[row]. All hot data is L2-resident.
// ---------------------------------------------------------------------------
__global__ __launch_bounds__(256)
void k_prop(const float4* __restrict__ x, float* __restrict__ y,
            const int* __restrict__ row, const int* __restrict__ col,
            const float* __restrict__ w) {
  __shared__ int smem[TILE * 3];                 // [0,TILE): col  [TILE,2T): row  [2T,3T): w
  int*   s_col = smem;
  int*   s_row = smem + TILE;
  float* s_w   = (float*)(smem + 2 * TILE);

  const unsigned base      = (unsigned)blockIdx.x * TILE;
  const unsigned remaining = (unsigned)N_EDGES - base;
  const unsigned cnt       = remaining < (unsigned)TILE ? remaining : (unsigned)TILE;

  if (threadIdx.x < 32) {                        // wave 0 issues the 3 DMA descriptors
    const unsigned lds_base = (unsigned)(uintptr_t)(void*)smem;  // LDS aperture: low 32 bits = byte offset
    tdm_load_1d_b32(lds_base,            (const void*)(col + base), remaining, TILE);
    tdm_load_1d_b32(lds_base + TILE * 4, (const void*)(row + base), remaining, TILE);
    tdm_load_1d_b32(lds_base + TILE * 8, (const void*)(w   + base), remaining, TILE);
    __builtin_amdgcn_s_wait_tensorcnt(0);
  }
  __syncthreads();

  const unsigned grp  = threadIdx.x >> 4;        // 16 groups of 16 lanes
  const unsigned lane = threadIdx.x & 15;        // float4 chunk within the 64-dim row

  for (unsigned j = grp; j < cnt; j += 16) {
    const int   c  = s_col[j];
    const int   r  = s_row[j];
    const float wt = s_w[j];
    const float4 v = x[(unsigned)c * 16u + lane];          // 128-bit gather
    float* dst = y + (unsigned)r * 64u + lane * 4u;
    __hip_atomic_fetch_add(dst + 0, wt * v.x, __ATOMIC_RELAXED, __HIP_MEMORY_SCOPE_AGENT);
    __hip_atomic_fetch_add(dst + 1, wt * v.y, __ATOMIC_RELAXED, __HIP_MEMORY_SCOPE_AGENT);
    __hip_atomic_fetch_add(dst + 2, wt * v.z, __ATOMIC_RELAXED, __HIP_MEMORY_SCOPE_AGENT);
    __hip_atomic_fetch_add(dst + 3, wt * v.w, __ATOMIC_RELAXED, __HIP_MEMORY_SCOPE_AGENT);
  }
}

// out = xcur = emb ; xnext = 0
__global__ __launch_bounds__(256)
void k_init(const float4* __restrict__ emb, float4* __restrict__ out,
            float4* __restrict__ xcur, float4* __restrict__ xnext, int n4) {
  int i = blockIdx.x * blockDim.x + threadIdx.x;
  if (i < n4) {
    float4 e = emb[i];
    out[i]  = e;
    xcur[i] = e;
    float4 zz; zz.x = 0.f; zz.y = 0.f; zz.z = 0.f; zz.w = 0.f;
    xnext[i] = zz;
  }
}

// out = (out + xnext) * scale ; xcur = xnext ; xnext = 0
__global__ __launch_bounds__(256)
void k_acc(float4* __restrict__ out, float4* __restrict__ xcur,
           float4* __restrict__ xnext, float scale, int n4) {
  int i = blockIdx.x * blockDim.x + threadIdx.x;
  if (i < n4) {
    float4 o  = out[i];
    float4 xn = xnext[i];
    o.x = (o.x + xn.x) * scale;
    o.y = (o.y + xn.y) * scale;
    o.z = (o.z + xn.z) * scale;
    o.w = (o.w + xn.w) * scale;
    out[i]  = o;
    xcur[i] = xn;
    float4 zz; zz.x = 0.f; zz.y = 0.f; zz.z = 0.f; zz.w = 0.f;
    xnext[i] = zz;
  }
}

extern "C" void kernel_launch(void* const* d_in, const int* in_sizes, int n_in,
                              void* d_out, int out_size, void* d_ws, size_t ws_size,
                              hipStream_t stream) {
  const float* emb = (const float*)d_in[0];   // (NUM_NODES, 64) f32
  const float* ew  = (const float*)d_in[1];   // (N_EDGES,)      f32
  const int*   row = (const int*)d_in[2];     // (N_EDGES,)      i32
  const int*   col = (const int*)d_in[3];     // (N_EDGES,)      i32

  float* out   = (float*)d_out;               // NUM_NODES*64 f32 (user||item flat)
  float* xcur  = (float*)d_ws;                // 38.4 MB
  float* xnext = xcur + (size_t)NUM_NODES * EMBED_DIM;  // 38.4 MB

  const int n4  = NUM_NODES * EMBED_DIM / 4;  // 2.4M float4
  const int ewb = (n4 + 255) / 256;
  const int epb = (N_EDGES + TILE - 1) / TILE;

  k_init<<<ewb, 256, 0, stream>>>((const float4*)emb, (float4*)out,
                                  (float4*)xcur, (float4*)xnext, n4);

  for (int layer = 0; layer < N_LAYERS; ++layer) {
    k_prop<<<epb, 256, 0, stream>>>((const float4*)xcur, xnext, row, col, ew);
    const float scale = (layer == N_LAYERS - 1) ? (1.0f / (N_LAYERS + 1)) : 1.0f;
    k_acc<<<ewb, 256, 0, stream>>>((float4*)out, (float4*)xcur, (float4*)xnext,
                                   scale, n4);
  }
}